// QuantumStateSuperposition_89154931130740
// MI455X (gfx1250) — compile-verified
//
#include <hip/hip_runtime.h>
#include <hip/hip_bf16.h>
#include <stdint.h>

// Shapes: B=512, S=64, D=2048, P=D/2=1024
#define B_ 512
#define S_ 64
#define D_ 2048
#define P_ 1024

typedef float v2f __attribute__((ext_vector_type(2)));
typedef float v8f __attribute__((ext_vector_type(8)));

// ---------------------------------------------------------------------------
// Kernel 1: per-row prep.
//   block t (64 blocks): row-norm of amplitudes, real_amp = amp/||.|| * (cos+sin)(ph)
//   M0[t,d] = real_amp[t,d] * g[t,0,d&1]; M1[t,d] = real_amp[t,d] * g[t,1,d&1]
//   entW[t,:] = softmax(E[t,:])
// ---------------------------------------------------------------------------
__global__ void k_prep(const float* __restrict__ amp, const float* __restrict__ ph,
                       const float* __restrict__ rot, const float* __restrict__ E,
                       float* __restrict__ entW, float* __restrict__ M0,
                       float* __restrict__ M1) {
  const int t = blockIdx.x;
  const int tid = threadIdx.x;
  __shared__ float sred[256];
  __shared__ float srow[64];
  __shared__ float smax, ssum;

  const float* arow = amp + t * D_;
  const float* prow = ph + t * D_;

  float ss = 0.f;
  for (int d = tid; d < D_; d += 256) { float a = arow[d]; ss += a * a; }
  sred[tid] = ss;
  __syncthreads();
  for (int off = 128; off > 0; off >>= 1) {
    if (tid < off) sred[tid] += sred[tid + off];
    __syncthreads();
  }
  const float inv = 1.0f / sqrtf(sred[0]);

  const float g00 = rot[t * 4 + 0], g01 = rot[t * 4 + 1];
  const float g10 = rot[t * 4 + 2], g11 = rot[t * 4 + 3];
  for (int d = tid; d < D_; d += 256) {
    float s, c;
    sincosf(prow[d], &s, &c);
    const float ra = arow[d] * inv * (c + s);
    const int j = d & 1;
    M0[t * D_ + d] = ra * (j ? g01 : g00);
    M1[t * D_ + d] = ra * (j ? g11 : g10);
  }

  // softmax of row t of entanglement matrix
  if (tid < 64) srow[tid] = E[t * 64 + tid];
  __syncthreads();
  if (tid == 0) {
    float m = srow[0];
    for (int i = 1; i < 64; ++i) m = fmaxf(m, srow[i]);
    smax = m;
  }
  __syncthreads();
  if (tid < 64) srow[tid] = expf(srow[tid] - smax);
  __syncthreads();
  if (tid == 0) {
    float s2 = 0.f;
    for (int i = 0; i < 64; ++i) s2 += srow[i];
    ssum = s2;
  }
  __syncthreads();
  if (tid < 64) entW[t * 64 + tid] = srow[tid] / ssum;
}

// ---------------------------------------------------------------------------
// Kernel 2: Gumbel noise exactly matching jax.random.categorical(key(42), ...)
// threefry2x32 with key (0, 42); counts = iota(32768) split as (i, i+16384).
// ---------------------------------------------------------------------------
__device__ __forceinline__ uint32_t tf_rotl(uint32_t x, int d) {
  return (x << d) | (x >> (32 - d));
}

__device__ __forceinline__ void threefry2x32(uint32_t k0, uint32_t k1,
                                             uint32_t& x0, uint32_t& x1) {
  const uint32_t ks2 = k0 ^ k1 ^ 0x1BD11BDAu;
  const int ra[4] = {13, 15, 26, 6};
  const int rb[4] = {17, 29, 16, 24};
  x0 += k0; x1 += k1;
  for (int r = 0; r < 4; ++r) { x0 += x1; x1 = tf_rotl(x1, ra[r]); x1 ^= x0; }
  x0 += k1; x1 += ks2 + 1u;
  for (int r = 0; r < 4; ++r) { x0 += x1; x1 = tf_rotl(x1, rb[r]); x1 ^= x0; }
  x0 += ks2; x1 += k0 + 2u;
  for (int r = 0; r < 4; ++r) { x0 += x1; x1 = tf_rotl(x1, ra[r]); x1 ^= x0; }
  x0 += k0; x1 += k1 + 3u;
  for (int r = 0; r < 4; ++r) { x0 += x1; x1 = tf_rotl(x1, rb[r]); x1 ^= x0; }
  x0 += k1; x1 += ks2 + 4u;
  for (int r = 0; r < 4; ++r) { x0 += x1; x1 = tf_rotl(x1, ra[r]); x1 ^= x0; }
  x0 += ks2; x1 += k0 + 5u;
}

__device__ __forceinline__ float gumbel_from_bits(uint32_t bits) {
  const float TINY = 1.17549435e-38f;
  float f = __uint_as_float((bits >> 9) | 0x3F800000u) - 1.0f;  // [0,1)
  float u = fmaxf(TINY, f * (1.0f - TINY) + TINY);
  return -logf(-logf(u));
}

__global__ void k_gumbel(float* __restrict__ g) {
  const uint32_t n = blockIdx.x * blockDim.x + threadIdx.x;  // 0..16383
  uint32_t x0 = n, x1 = n + 16384u;
  threefry2x32(0u, 42u, x0, x1);
  g[n] = gumbel_from_bits(x0);
  g[n + 16384] = gumbel_from_bits(x1);
}

// ---------------------------------------------------------------------------
// Kernel 3: A_i = entW(64x64) @ M_i(64x2048) via V_WMMA_F32_16X16X4_F32.
// 1024 output tiles (2 matrices x 4 m-tiles x 128 n-tiles), one per wave.
// Fragment layout (ISA 7.12.2): A lane<16 -> K=k,k+1 ; lane>=16 -> K=k+2,k+3.
// B VGPR0/1 -> rows {k,k+1} (lanes 0-15) / {k+2,k+3} (lanes 16-31).
// D VGPR r -> row m0+r (lanes 0-15) / m0+r+8 (lanes 16-31).
// ---------------------------------------------------------------------------
__global__ void k_agemm(const float* __restrict__ entW, const float* __restrict__ M0,
                        const float* __restrict__ M1, float* __restrict__ A0,
                        float* __restrict__ A1) {
  const int wave = threadIdx.x >> 5;
  const int lane = threadIdx.x & 31;
  const int gw = blockIdx.x * 4 + wave;       // 0..1023
  const int mat = gw >> 9;                    // 0 or 1
  const int r = gw & 511;
  const int m0 = (r >> 7) * 16;
  const int n0 = (r & 127) * 16;
  const float* __restrict__ M = mat ? M1 : M0;
  float* __restrict__ A = mat ? A1 : A0;
  const int lanelo = lane & 15;
  const int half = lane >> 4;
  const int m = m0 + lanelo;
  const int n = n0 + lanelo;

  v8f c = {};
  for (int k = 0; k < 64; k += 4) {
    const int kk = k + 2 * half;
    v2f a, b;
    a.x = entW[m * 64 + kk];
    a.y = entW[m * 64 + kk + 1];
    b.x = M[kk * D_ + n];
    b.y = M[(kk + 1) * D_ + n];
    c = __builtin_amdgcn_wmma_f32_16x16x4_f32(false, a, false, b, (short)0, c,
                                              false, false);
  }
  for (int rr = 0; rr < 8; ++rr)
    A[(m0 + rr + 8 * half) * D_ + n] = c[rr];
}

// ---------------------------------------------------------------------------
// Kernel 4: logits partials via WMMA. Conceptual GEMM: U[B, 4P] @ C[4P, S]
// where per pair p:  U[b, 4p+*] = (x0^2, x0*x1, x1^2, 0)
//                    C[4p+*, s] = (A0e^2+A0o^2, 2(A0.A1), A1e^2+A1o^2, 0)
// One k-step-of-4 == one pair; fragments built on the fly (L2-resident data).
// K split into 8 chunks of 128 pairs -> deterministic partial buffer.
// Grid: 256 blocks x 128 thr; block = (b_tile 0..31, kc 0..7), wave -> s_tile.
// ---------------------------------------------------------------------------
__global__ void k_logits(const float* __restrict__ x, const float* __restrict__ A0,
                         const float* __restrict__ A1, float* __restrict__ part) {
  const int wave = threadIdx.x >> 5;
  const int lane = threadIdx.x & 31;
  const int bt = blockIdx.x & 31;
  const int kc = blockIdx.x >> 5;
  const int b0 = bt * 16;
  const int s0 = wave * 16;
  const int lanelo = lane & 15;
  const int half = lane >> 4;

  const float2* __restrict__ xr = (const float2*)(x + (b0 + lanelo) * D_);
  const float2* __restrict__ a0r = (const float2*)(A0 + (s0 + lanelo) * D_);
  const float2* __restrict__ a1r = (const float2*)(A1 + (s0 + lanelo) * D_);

  v8f c = {};
  const int pbeg = kc * 128, pend = pbeg + 128;
  for (int p = pbeg; p < pend; ++p) {
    const float2 xv = xr[p];
    const float2 a0 = a0r[p];
    const float2 a1 = a1r[p];
    v2f a, b;
    a.x = half ? xv.y * xv.y : xv.x * xv.x;
    a.y = half ? 0.0f : xv.x * xv.y;
    b.x = half ? (a1.x * a1.x + a1.y * a1.y) : (a0.x * a0.x + a0.y * a0.y);
    b.y = half ? 0.0f : 2.0f * (a0.x * a1.x + a0.y * a1.y);
    c = __builtin_amdgcn_wmma_f32_16x16x4_f32(false, a, false, b, (short)0, c,
                                              false, false);
  }
  float* __restrict__ dst = part + (size_t)kc * B_ * S_;
  for (int rr = 0; rr < 8; ++rr)
    dst[(b0 + rr + 8 * half) * S_ + s0 + lanelo] = c[rr];
}

// ---------------------------------------------------------------------------
// Kernel 5: per-b reduce partials, + gumbel, argmax (first-index tie-break),
// then collapsed[b,d] = x0*A0[o,d] + x1*A1[o,d].
// ---------------------------------------------------------------------------
__global__ void k_collapse(const float* __restrict__ x, const float* __restrict__ A0,
                           const float* __restrict__ A1, const float* __restrict__ part,
                           const float* __restrict__ gum, float* __restrict__ out) {
  const int b = blockIdx.x;
  const int tid = threadIdx.x;
  __shared__ float sval[64];
  __shared__ int sidx[64];

  if (tid < 64) {
    float l = 0.f;
    for (int kc = 0; kc < 8; ++kc)
      l += part[((size_t)kc * B_ + b) * S_ + tid];
    sval[tid] = l + gum[b * 64 + tid];
    sidx[tid] = tid;
  }
  __syncthreads();
  for (int off = 32; off > 0; off >>= 1) {
    if (tid < off) {
      const float v2 = sval[tid + off];
      const int i2 = sidx[tid + off];
      if (v2 > sval[tid] || (v2 == sval[tid] && i2 < sidx[tid])) {
        sval[tid] = v2;
        sidx[tid] = i2;
      }
    }
    __syncthreads();
  }
  const int o = sidx[0];

  const float2* __restrict__ xr = (const float2*)(x + b * D_);
  const float2* __restrict__ a0r = (const float2*)(A0 + o * D_);
  const float2* __restrict__ a1r = (const float2*)(A1 + o * D_);
  float2* __restrict__ outr = (float2*)(out + b * D_);
  for (int p = tid; p < P_; p += 256) {
    const float2 xv = xr[p];
    const float2 a0 = a0r[p];
    const float2 a1 = a1r[p];
    float2 rr;
    rr.x = xv.x * a0.x + xv.y * a1.x;
    rr.y = xv.x * a0.y + xv.y * a1.y;
    outr[p] = rr;
  }
}

// ---------------------------------------------------------------------------
extern "C" void kernel_launch(void* const* d_in, const int* in_sizes, int n_in,
                              void* d_out, int out_size, void* d_ws, size_t ws_size,
                              hipStream_t stream) {
  const float* x   = (const float*)d_in[0];  // classical_state [512,2048]
  const float* amp = (const float*)d_in[1];  // amplitudes [64,2048]
  const float* ph  = (const float*)d_in[2];  // phases [64,2048]
  const float* rot = (const float*)d_in[3];  // rotation_gates [64,2,2]
  const float* E   = (const float*)d_in[4];  // entanglement_matrix [64,64]
  float* out = (float*)d_out;                // [512,2048]

  // Workspace layout (floats): total 823,296 floats = ~3.3 MB
  float* ws = (float*)d_ws;
  float* entW = ws;                    //  4096
  float* M0   = entW + 64 * 64;        //  131072
  float* M1   = M0 + S_ * D_;          //  131072
  float* A0   = M1 + S_ * D_;          //  131072
  float* A1   = A0 + S_ * D_;          //  131072
  float* gum  = A1 + S_ * D_;          //  32768
  float* part = gum + B_ * S_;         //  262144 (8 x 512 x 64)

  k_prep<<<64, 256, 0, stream>>>(amp, ph, rot, E, entW, M0, M1);
  k_gumbel<<<64, 256, 0, stream>>>(gum);
  k_agemm<<<256, 128, 0, stream>>>(entW, M0, M1, A0, A1);
  k_logits<<<256, 128, 0, stream>>>(x, A0, A1, part);
  k_collapse<<<512, 256, 0, stream>>>(x, A0, A1, part, gum, out);
}